// UnidirectionalAdjacencyControl_22273700397207
// MI455X (gfx1250) — compile-verified
//
#include <hip/hip_runtime.h>
#include <stdint.h>

typedef __attribute__((ext_vector_type(2))) float v2f;
typedef __attribute__((ext_vector_type(8))) float v8f;

#define N_NODES 8192
#define F_DIM   256
#define N_EDGES 262144

// ---- zero D, c, and the argmax slot -------------------------------------
__global__ void k_init(uint32_t* __restrict__ D, uint32_t* __restrict__ cnt,
                       unsigned long long* __restrict__ slot) {
  int i = blockIdx.x * blockDim.x + threadIdx.x;
  if (i < N_NODES) { D[i] = 0u; cnt[i] = 0u; }
  if (i == 0) *slot = 0ull;
}

// ---- D = histogram of edge sources (duplicates accumulate) --------------
__global__ void k_hist(const int* __restrict__ src, uint32_t* __restrict__ D) {
  int e = blockIdx.x * blockDim.x + threadIdx.x;
  if (e < N_EDGES) atomicAdd(&D[(uint32_t)src[e]], 1u);
}

// ---- argmax(D) with smallest-index tiebreak via packed 64-bit atomicMax -
__global__ void k_argmax(const uint32_t* __restrict__ D,
                         unsigned long long* __restrict__ slot) {
  int i = blockIdx.x * blockDim.x + threadIdx.x;
  if (i < N_NODES) {
    unsigned long long key =
        ((unsigned long long)D[i] << 32) |
        (unsigned long long)(0xFFFFFFFFu - (uint32_t)i);
    atomicMax(slot, key);
  }
}

// ---- c[i] = #edges (i -> top) -------------------------------------------
__global__ void k_colcount(const int* __restrict__ src, const int* __restrict__ dst,
                           const unsigned long long* __restrict__ slot,
                           uint32_t* __restrict__ cnt) {
  uint32_t top = 0xFFFFFFFFu - (uint32_t)(*slot);
  int e = blockIdx.x * blockDim.x + threadIdx.x;
  if (e < N_EDGES && (uint32_t)dst[e] == top)
    atomicAdd(&cnt[(uint32_t)src[e]], 1u);
}

// ---- h_top = W @ x[top] + b via V_WMMA_F32_16X16X4_F32 ------------------
// One block, 512 threads = 16 waves. Wave w computes output cols [16w,16w+16)
// for the 16 rows [top&~15, (top&~15)+16); each wave loops K in steps of 4.
// h[m,n] = sum_k x[r0+m,k] * W[n,k]  (B = W^T so B[k][n] = W[n][k]).
__global__ void __launch_bounds__(512)
k_gemv_wmma(const float* __restrict__ x, const float* __restrict__ W,
            const float* __restrict__ bias,
            const unsigned long long* __restrict__ slot,
            float* __restrict__ h_top) {
  const uint32_t top = 0xFFFFFFFFu - (uint32_t)(*slot);
  const uint32_t r0  = top & ~15u;

  const int lane = threadIdx.x & 31;
  const int wave = threadIdx.x >> 5;
  const int n0   = wave << 4;          // output column tile
  const int half = lane >> 4;          // 0: lanes 0-15, 1: lanes 16-31
  const int l    = lane & 15;

  v8f acc = {};
  const float* xrow = x + (size_t)(r0 + (uint32_t)l) * F_DIM;   // A row m=l
  const float* wrow = W + (size_t)(n0 + l) * F_DIM;             // B col n=l

  for (int k0 = 0; k0 < F_DIM; k0 += 4) {
    // A 16x4 f32 layout: lanes 0-15 hold (K=0,K=1), lanes 16-31 hold (K=2,K=3)
    v2f a;
    a.x = xrow[k0 + half * 2 + 0];
    a.y = xrow[k0 + half * 2 + 1];
    // B 4x16 f32: rows striped across lanes; VGPR0 rows K=0/K=1, VGPR1 rows K=2/K=3
    v2f bm;
    bm.x = wrow[k0 + half + 0];
    bm.y = wrow[k0 + half + 2];
    acc = __builtin_amdgcn_wmma_f32_16x16x4_f32(
        /*neg_a=*/false, a, /*neg_b=*/false, bm,
        /*c_mod=*/(short)0, acc, /*reuse_a=*/false, /*reuse_b=*/false);
  }

  // Extract row m = top & 15: lanes 0-15 hold M=0..7 (VGPR m), lanes 16-31 hold M=8..15
  const uint32_t m = top & 15u;
  if ((int)(m >> 3) == half) {
    int n = n0 + l;
    float v;
    switch (m & 7u) {
      case 0: v = acc[0]; break;
      case 1: v = acc[1]; break;
      case 2: v = acc[2]; break;
      case 3: v = acc[3]; break;
      case 4: v = acc[4]; break;
      case 5: v = acc[5]; break;
      case 6: v = acc[6]; break;
      default: v = acc[7]; break;
    }
    h_top[n] = v + bias[n];
  }
}

// ---- out[i,:] = c[i] * h_top[:], vectorized float4 writes ---------------
__global__ void k_broadcast(const uint32_t* __restrict__ cnt,
                            const float* __restrict__ h_top,
                            float* __restrict__ out) {
  int t  = blockIdx.x * blockDim.x + threadIdx.x;   // 0 .. N*F/4-1
  int i  = t >> 6;                                  // 64 float4 per row
  int f4 = t & 63;
  float s = (float)cnt[i];
  float4 hv = ((const float4*)h_top)[f4];
  float4 o;
  o.x = s * hv.x; o.y = s * hv.y; o.z = s * hv.z; o.w = s * hv.w;
  ((float4*)out)[t] = o;
}

extern "C" void kernel_launch(void* const* d_in, const int* in_sizes, int n_in,
                              void* d_out, int out_size, void* d_ws, size_t ws_size,
                              hipStream_t stream) {
  const float* x    = (const float*)d_in[0];   // (N, F)
  const float* W    = (const float*)d_in[1];   // (F, F)
  const float* bias = (const float*)d_in[2];   // (F,)
  const int*   edge = (const int*)d_in[3];     // (2, E) flat
  // d_in[4] = batch_index: unused by the reference output (bug preserved)

  const int* src = edge;            // edge_index[0]
  const int* dst = edge + N_EDGES;  // edge_index[1]

  uint32_t* D   = (uint32_t*)d_ws;                                   // N u32
  uint32_t* cnt = D + N_NODES;                                       // N u32
  unsigned long long* slot = (unsigned long long*)(D + 2 * N_NODES); // 1 u64
  float* h_top = (float*)((char*)d_ws + (size_t)2 * N_NODES * 4 + 64);

  float* out = (float*)d_out;

  k_init<<<(N_NODES + 255) / 256, 256, 0, stream>>>(D, cnt, slot);
  k_hist<<<N_EDGES / 256, 256, 0, stream>>>(src, D);
  k_argmax<<<(N_NODES + 255) / 256, 256, 0, stream>>>(D, slot);
  k_colcount<<<N_EDGES / 256, 256, 0, stream>>>(src, dst, slot, cnt);
  k_gemv_wmma<<<1, 512, 0, stream>>>(x, W, bias, slot, h_top);
  k_broadcast<<<(N_NODES * F_DIM / 4) / 256, 256, 0, stream>>>(cnt, h_top, out);
}